// GATv2AttentionHead_38835094290621
// MI455X (gfx1250) — compile-verified
//
#include <hip/hip_runtime.h>

typedef __attribute__((ext_vector_type(2))) float v2f;
typedef __attribute__((ext_vector_type(8))) float v8f;

#define DIN   128
#define DOUT  32
#define SLOPE 0.2f

// ---------------------------------------------------------------------------
// Kernel 1: h = x @ W1^T + b   via V_WMMA_F32_16X16X4_F32 (exact fp32)
// One wave computes a 16(row) x 32(col) tile: two 16x16 accumulators,
// K advanced 4 per WMMA. W1 (32x128 = 16KB) staged in LDS.
// ---------------------------------------------------------------------------
__global__ __launch_bounds__(256) void gat_gemm_wmma(
    const float* __restrict__ x, const float* __restrict__ W,
    const float* __restrict__ bias, float* __restrict__ h, int N)
{
    __shared__ float lW[DOUT * DIN];        // W[n][k], row-major, 16 KB
    const int tid = threadIdx.x;
    {   // cooperative stage of W into LDS (float4)
        const float4* s4 = (const float4*)W;
        float4*       d4 = (float4*)lW;
        for (int i = tid; i < DOUT * DIN / 4; i += 256) d4[i] = s4[i];
    }
    __syncthreads();

    const int wave  = tid >> 5;             // 8 waves/block -> 128 rows/block
    const int lane  = tid & 31;
    const int half  = lane >> 4;            // K-pair selector (ISA A/B layout)
    const int lm    = lane & 15;
    const int mbase = blockIdx.x * 128 + wave * 16;
    if (mbase >= N) return;                 // wave-uniform: EXEC stays all-ones

    int row = mbase + lm; if (row > N - 1) row = N - 1;   // clamp tail reads
    const float* xrow = x + (long)row * DIN;

    v8f acc0 = {}; v8f acc1 = {};
    #pragma unroll
    for (int kk = 0; kk < DIN; kk += 4) {
        const int k = kk + 2 * half;        // lanes 0-15: K=kk,kk+1; 16-31: kk+2,kk+3
        v2f a  = *(const v2f*)(xrow + k);                  // A[16x4] fragment
        v2f b0 = *(const v2f*)(lW + lm * DIN + k);         // B[4x16] cols 0..15
        v2f b1 = *(const v2f*)(lW + (16 + lm) * DIN + k);  // B[4x16] cols 16..31
        acc0 = __builtin_amdgcn_wmma_f32_16x16x4_f32(false, a, false, b0,
                                                     (short)0, acc0, false, false);
        acc1 = __builtin_amdgcn_wmma_f32_16x16x4_f32(false, a, false, b1,
                                                     (short)0, acc1, false, false);
    }

    // C/D layout: VGPR r, lanes 0-15 -> M=r, N=lane; lanes 16-31 -> M=8+r, N=lane-16
    const float b0v = bias[lm], b1v = bias[16 + lm];
    #pragma unroll
    for (int r = 0; r < 8; ++r) {
        const int m = mbase + r + 8 * half;
        if (m < N) {
            h[(long)m * DOUT + lm]      = acc0[r] + b0v;
            h[(long)m * DOUT + 16 + lm] = acc1[r] + b1v;
        }
    }
}

// ---------------------------------------------------------------------------
// Kernel 2: per-node scores s1 = leakyrelu(h).a1, s2 = leakyrelu(h).a2
// One wave per node (32 lanes == 32 channels), butterfly reduction.
// ---------------------------------------------------------------------------
__global__ __launch_bounds__(256) void gat_scores(
    const float* __restrict__ h, const float* __restrict__ a1,
    const float* __restrict__ a2, float* __restrict__ s1,
    float* __restrict__ s2, int N)
{
    const int node = blockIdx.x * 8 + (threadIdx.x >> 5);
    const int c    = threadIdx.x & 31;
    if (node >= N) return;
    const float hv = h[(long)node * DOUT + c];
    const float z  = hv > 0.f ? hv : SLOPE * hv;
    float p1 = z * a1[c];
    float p2 = z * a2[c];
    #pragma unroll
    for (int m = 16; m >= 1; m >>= 1) {
        p1 += __shfl_xor(p1, m, 32);
        p2 += __shfl_xor(p2, m, 32);
    }
    if (c == 0) { s1[node] = p1; s2[node] = p2; }
}

// order-preserving float<->u32 key for atomic max
__device__ __forceinline__ unsigned fenc(float f) {
    unsigned b = __float_as_uint(f);
    return (b & 0x80000000u) ? ~b : (b | 0x80000000u);
}
__device__ __forceinline__ float fdec(unsigned k) {
    return __uint_as_float((k & 0x80000000u) ? (k & 0x7fffffffu) : ~k);
}

// ---------------------------------------------------------------------------
// Kernel 3: init accumulators (harness poisons d_out/d_ws with 0xAA)
// ---------------------------------------------------------------------------
__global__ void gat_init(float* __restrict__ out, unsigned* __restrict__ keys,
                         float* __restrict__ denom, int N)
{
    const long i = (long)blockIdx.x * 256 + threadIdx.x;
    if (i < (long)N * DOUT) out[i] = 0.f;
    if (i < N) { keys[i] = 0u; denom[i] = 0.f; }   // key 0 < fenc(any real float)
}

// ---------------------------------------------------------------------------
// Kernel 4: edge pass A — logits + segment max over src (L2-resident atomics)
// ---------------------------------------------------------------------------
__global__ void gat_edge_max(const int* __restrict__ src, const int* __restrict__ dst,
                             const float* __restrict__ s1, const float* __restrict__ s2,
                             float* __restrict__ logits, unsigned* __restrict__ keys,
                             int E, int N)
{
    const long t = (long)blockIdx.x * 256 + threadIdx.x;
    if (t >= (long)E + N) return;
    const int s = (t < E) ? src[t] : (int)(t - E);   // self-loops appended
    const int d = (t < E) ? dst[t] : (int)(t - E);
    const float lg = s1[s] + s2[d];
    logits[t] = lg;
    atomicMax(&keys[s], fenc(lg));
}

// ---------------------------------------------------------------------------
// Kernel 5: edge pass B — ex = exp(logit - max); denom[src]+=ex;
//           out[src] += ex * h[dst]   (normalize later: alpha = ex/denom)
// ---------------------------------------------------------------------------
__global__ void gat_edge_acc(const int* __restrict__ src, const int* __restrict__ dst,
                             const float* __restrict__ logits,
                             const unsigned* __restrict__ keys,
                             const float* __restrict__ h,
                             float* __restrict__ denom, float* __restrict__ out,
                             int E, int N)
{
    const long t = (long)blockIdx.x * 256 + threadIdx.x;
    if (t >= (long)E + N) return;
    const int s = (t < E) ? src[t] : (int)(t - E);
    const int d = (t < E) ? dst[t] : (int)(t - E);
    const float ex = __expf(logits[t] - fdec(keys[s]));
    atomicAdd(&denom[s], ex);
    const float* hd = h + (long)d * DOUT;
    float*       os = out + (long)s * DOUT;
    #pragma unroll
    for (int c = 0; c < DOUT; c += 4) {
        const float4 hv = *(const float4*)(hd + c);   // 128B L2-resident gather
        atomicAdd(os + c + 0, ex * hv.x);
        atomicAdd(os + c + 1, ex * hv.y);
        atomicAdd(os + c + 2, ex * hv.z);
        atomicAdd(os + c + 3, ex * hv.w);
    }
}

// ---------------------------------------------------------------------------
// Kernel 6: out[i][c] /= denom[i]
// ---------------------------------------------------------------------------
__global__ void gat_norm(float* __restrict__ out, const float* __restrict__ denom, int N)
{
    const long i = (long)blockIdx.x * 256 + threadIdx.x;
    if (i < (long)N * DOUT) out[i] /= denom[i / DOUT];
}

// ---------------------------------------------------------------------------
extern "C" void kernel_launch(void* const* d_in, const int* in_sizes, int n_in,
                              void* d_out, int out_size, void* d_ws, size_t ws_size,
                              hipStream_t stream)
{
    const float* x  = (const float*)d_in[0];   // [N,128]
    const int*   ei = (const int*)  d_in[1];   // [2,E]
    const float* W  = (const float*)d_in[2];   // [32,128]
    const float* b  = (const float*)d_in[3];   // [32]
    const float* a1 = (const float*)d_in[4];   // [32]
    const float* a2 = (const float*)d_in[5];   // [32]

    const int N = in_sizes[0] / DIN;
    const int E = in_sizes[1] / 2;
    const int* src = ei;
    const int* dst = ei + E;

    // workspace layout (floats): h[N*32] | s1[N] | s2[N] | keys[N] | denom[N] | logits[E+N]
    float*    ws     = (float*)d_ws;
    float*    h      = ws;
    float*    s1     = h + (size_t)N * DOUT;
    float*    s2     = s1 + N;
    unsigned* keys   = (unsigned*)(s2 + N);
    float*    denom  = (float*)(keys + N);
    float*    logits = denom + N;
    float*    out    = (float*)d_out;

    const long M  = (long)E + N;            // edges incl. self-loops
    const long NC = (long)N * DOUT;

    gat_gemm_wmma<<<(N + 127) / 128, 256, 0, stream>>>(x, W, b, h, N);
    gat_scores  <<<(N + 7) / 8,     256, 0, stream>>>(h, a1, a2, s1, s2, N);
    gat_init    <<<(unsigned)((NC + 255) / 256), 256, 0, stream>>>(out, keys, denom, N);
    gat_edge_max<<<(unsigned)((M + 255) / 256),  256, 0, stream>>>(src, dst, s1, s2, logits, keys, E, N);
    gat_edge_acc<<<(unsigned)((M + 255) / 256),  256, 0, stream>>>(src, dst, logits, keys, h, denom, out, E, N);
    gat_norm    <<<(unsigned)((NC + 255) / 256), 256, 0, stream>>>(out, denom, N);
}